// RSMLayer_29274497090009
// MI455X (gfx1250) — compile-verified
//
#include <hip/hip_runtime.h>
#include <math.h>

// Problem constants from the reference
#define B_T   256      // BATCH
#define D_IN  784
#define M_G   800      // M (groups)
#define N_G   8        // N (group width)
#define MN    6400     // M*N
#define K_TOP 25
#define TPB   1024

typedef float v2f __attribute__((ext_vector_type(2)));
typedef float v8f __attribute__((ext_vector_type(8)));

// ---------------------------------------------------------------------------
// Kernel 1: A_all[t][m] = sum_d X[t][d] * Wa[m][d] + ba[m]
// One wave (32 threads) per 16x16 tile of the (256 x 800) output.
// Uses V_WMMA_F32_16X16X4_F32: A = 16(t) x 4(d), B = 4(d) x 16(m).
// ---------------------------------------------------------------------------
__global__ __launch_bounds__(32) void rsm_gemm_a_wmma(
    const float* __restrict__ X,    // [256,784]
    const float* __restrict__ Wa,   // [800,784]
    const float* __restrict__ ba,   // [800]
    float* __restrict__ Aall)       // [256,800]
{
    const int lane = threadIdx.x & 31;
    const int tile = blockIdx.x;            // 0..799
    const int tT = tile & 15;               // 16 tiles along batch (256/16)
    const int tM = tile >> 4;               // 50 tiles along M (800/16)

    const int lo16 = lane & 15;
    const int kb   = (lane >= 16) ? 2 : 0;  // K pair held by this half-wave

    const float* __restrict__ xrow = X  + (size_t)(tT * 16 + lo16) * D_IN; // A: row = t
    const float* __restrict__ wrow = Wa + (size_t)(tM * 16 + lo16) * D_IN; // B: col = m

    v8f acc = {};
    for (int k0 = 0; k0 < D_IN; k0 += 4) {
        v2f a, b;
        a.x = xrow[k0 + kb + 0];            // A lanes 0-15: K0,K1 ; lanes 16-31: K2,K3
        a.y = xrow[k0 + kb + 1];
        b.x = wrow[k0 + kb + 0];            // B VGPR0 = row K(kb), VGPR1 = row K(kb+1)
        b.y = wrow[k0 + kb + 1];
        acc = __builtin_amdgcn_wmma_f32_16x16x4_f32(
            /*neg_a=*/false, a, /*neg_b=*/false, b,
            /*c_mod=*/(short)0, acc, /*reuse_a=*/false, /*reuse_b=*/false);
    }

    // D layout: VGPR r -> M = r (+8 for lanes 16-31), N = lane%16
    const int   n    = lo16;
    const int   mhi  = (lane >= 16) ? 8 : 0;
    const float bias = ba[tM * 16 + n];
#pragma unroll
    for (int r = 0; r < 8; ++r) {
        const int trow = tT * 16 + r + mhi;
        Aall[(size_t)trow * M_G + tM * 16 + n] = acc[r] + bias;
    }
}

// ---------------------------------------------------------------------------
// Kernel 2: the 256-step sequential scan (single workgroup, 32 wave32s).
// Per-step state is provably sparse:
//   x_b  : one-hot of 1.0 at colIdx (or zero vector)  -> W_b@x_b = one column
//   phi  : single entry pval at phiIdx                 -> pi scalar correction
// ---------------------------------------------------------------------------
__global__ __launch_bounds__(TPB) void rsm_scan(
    const float* __restrict__ Aall,  // [256,800]  (precomputed a_t + b_a)
    const float* __restrict__ Wb,    // [6400,6400]
    const float* __restrict__ bb,    // [6400]
    const float* __restrict__ Wd,    // [784,800]
    const float* __restrict__ bd,    // [784]
    float* __restrict__ out)         // preds[256*784] | xb[6400] | phi[6400] | psi[6400]
{
    __shared__ float s_sigma[MN];
    __shared__ float s_lam[M_G];
    __shared__ float s_red[TPB];
    __shared__ int   s_redi[TPB];

    const int tid = threadIdx.x;

    int   colIdx = -1;   // x_b one-hot position (value 1.0), -1 => zero vector
    int   phiIdx = -1;   // phi single-entry position
    float pval   = 0.0f; // phi single-entry value

    for (int t = 0; t < B_T; ++t) {
        const float* __restrict__ arow = Aall + (size_t)t * M_G;

        // ---- sigma[k] = a[k % M] + b_b[k] + (colIdx>=0 ? Wb[k][colIdx] : 0)
        float lmin = INFINITY;
        for (int k = tid; k < MN; k += TPB) {
            float s = arow[k % M_G] + bb[k];
            if (colIdx >= 0) s += Wb[(size_t)k * MN + colIdx];
            s_sigma[k] = s;
            lmin = fminf(lmin, s);
        }
        // latency hint: pull next step's A row toward the caches
        if (t + 1 < B_T && tid < M_G)
            __builtin_prefetch(arow + M_G + tid, 0, 3);
        __syncthreads();

        // ---- min reduction over sigma
        s_red[tid] = lmin;
        __syncthreads();
        for (int off = TPB >> 1; off > 0; off >>= 1) {
            if (tid < off) s_red[tid] = fminf(s_red[tid], s_red[tid + off]);
            __syncthreads();
        }
        const float mnv = s_red[0];
        __syncthreads();   // everyone holds mnv before s_red is reused

        // ---- global argmax of pi (first-index tie-break, like jnp.argmax)
        float bv = -INFINITY; int bi = 0x7fffffff;
        for (int k = tid; k < MN; k += TPB) {
            const float f  = (k == phiIdx) ? (1.0f - pval) : 1.0f;
            const float pi = f * (s_sigma[k] - mnv + 1.0f);
            if (pi > bv) { bv = pi; bi = k; }   // ascending k => first max locally
        }
        s_red[tid] = bv; s_redi[tid] = bi;
        __syncthreads();
        for (int off = TPB >> 1; off > 0; off >>= 1) {
            if (tid < off) {
                const float ov = s_red[tid + off];
                const int   oi = s_redi[tid + off];
                if (ov > s_red[tid] || (ov == s_red[tid] && oi < s_redi[tid])) {
                    s_red[tid] = ov; s_redi[tid] = oi;
                }
            }
            __syncthreads();
        }
        const int kstar = s_redi[0];
        __syncthreads();   // everyone holds kstar before s_redi is reused

        // ---- lam[g] = max_j pi[g*8+j]  (consecutive-group max)
        if (tid < M_G) {
            float mx = -INFINITY;
#pragma unroll
            for (int j = 0; j < N_G; ++j) {
                const int   k  = tid * N_G + j;
                const float f  = (k == phiIdx) ? (1.0f - pval) : 1.0f;
                const float pi = f * (s_sigma[k] - mnv + 1.0f);
                mx = fmaxf(mx, pi);
            }
            s_lam[tid] = mx;
        }
        __syncthreads();

        // ---- is group g* = k* % M inside top-K of lam? (lax.top_k tie rule)
        const int   gstar = kstar % M_G;   // tiled layout: tile(m_lam,N)[k] = m_lam[k%M]
        const float Lg    = s_lam[gstar];
        int cnt = 0;
        if (tid < M_G) {
            const float lj = s_lam[tid];
            if (lj > Lg || (lj == Lg && tid < gstar)) cnt = 1;
        }
        s_redi[tid] = cnt;
        __syncthreads();
        for (int off = TPB >> 1; off > 0; off >>= 1) {
            if (tid < off) s_redi[tid] += s_redi[tid + off];
            __syncthreads();
        }
        const float mlam = (s_redi[0] < K_TOP) ? 1.0f : 0.0f;

        // ---- y is zero except y[k*] = tanh(mlam * sigma[k*])
        const float ystar = tanhf(mlam * s_sigma[kstar]);
        const float z     = fmaxf(0.0f, ystar);   // group-max of y at row i*
        const int   istar = kstar / N_G;

        // ---- preds[t] = b_d + z * W_d[:, i*]
        for (int d = tid; d < D_IN; d += TPB)
            out[(size_t)t * D_IN + d] = bd[d] + z * Wd[(size_t)d * M_G + istar];

        // ---- carry update (phi single-entry; x_b one-hot of 1.0 or zero)
        phiIdx = kstar;
        pval   = z;
        colIdx = (z > 0.0f) ? kstar : -1;

        __syncthreads();   // protect s_sigma/s_redi reads from next-iter writes
    }

    // ---- final sparse outputs: x_b, phi, psis[-1]
    float* __restrict__ xb  = out + (size_t)B_T * D_IN;
    float* __restrict__ phi = xb + MN;
    float* __restrict__ psi = phi + MN;
    for (int k = tid; k < MN; k += TPB) { xb[k] = 0.0f; phi[k] = 0.0f; psi[k] = 0.0f; }
    __threadfence_block();
    __syncthreads();
    if (tid == 0) {
        if (colIdx >= 0) xb[colIdx] = 1.0f;
        if (phiIdx >= 0) { phi[phiIdx] = pval; psi[phiIdx] = pval; }
    }
}

// ---------------------------------------------------------------------------
// Launch: inputs per setup_inputs() order:
//   0 batch_x [256,784]  1 W_a [800,784]  2 b_a [800]
//   3 W_b [6400,6400]    4 b_b [6400]     5 W_d [784,800]  6 b_d [784]
// Output: preds(200704) | x_b(6400) | phi(6400) | psis[-1](6400)
// ---------------------------------------------------------------------------
extern "C" void kernel_launch(void* const* d_in, const int* in_sizes, int n_in,
                              void* d_out, int out_size, void* d_ws, size_t ws_size,
                              hipStream_t stream) {
    const float* X  = (const float*)d_in[0];
    const float* Wa = (const float*)d_in[1];
    const float* ba = (const float*)d_in[2];
    const float* Wb = (const float*)d_in[3];
    const float* bb = (const float*)d_in[4];
    const float* Wd = (const float*)d_in[5];
    const float* bd = (const float*)d_in[6];
    float* out  = (float*)d_out;
    float* Aall = (float*)d_ws;   // 256*800*4 = 819,200 bytes of scratch

    // (256/16) * (800/16) = 800 tiles, one wave32 per tile
    rsm_gemm_a_wmma<<<dim3(800), dim3(32), 0, stream>>>(X, Wa, ba, Aall);

    // single-workgroup sequential scan (32 wave32s on one WGP)
    rsm_scan<<<dim3(1), dim3(TPB), 0, stream>>>(Aall, Wb, bb, Wd, bd, out);
}